// AdditiveAttention_12979391169303
// MI455X (gfx1250) — compile-verified
//
#include <hip/hip_runtime.h>
#include <hip/hip_bf16.h>

// Problem constants (match reference)
constexpr int BB = 16;    // batch
constexpr int QQ = 128;   // queries
constexpr int KK = 128;   // keys
constexpr int DD = 512;   // input dim
constexpr int HH = 512;   // hidden dim
constexpr float NEGBIG = -1e6f;
constexpr float LOG2E  = 1.4426950408889634f;

typedef __attribute__((ext_vector_type(2))) float v2f;
typedef __attribute__((ext_vector_type(8))) float v8f;

__device__ __forceinline__ float fast_tanh(float x) {
#if __has_builtin(__builtin_amdgcn_tanhf)
  return __builtin_amdgcn_tanhf(x);
#elif __has_builtin(__builtin_amdgcn_tanh_f32)
  return __builtin_amdgcn_tanh_f32(x);
#else
  // tanh(x) = (e^{2x}-1)/(e^{2x}+1), via v_exp_f32 (TRANS pipe). Clamp to avoid inf/inf.
  float cx = fminf(fmaxf(x, -15.f), 15.f);
  float e  = __builtin_amdgcn_exp2f(cx * 2.8853900817779268f); // 2*log2(e)
  return (e - 1.f) * __builtin_amdgcn_rcpf(e + 1.f);
#endif
}

// ---------------------------------------------------------------------------
// Kernel 1: fp32 WMMA GEMM  Out[M,H] = A[M,D] @ W[D,H]
// grid = (M/16, H/(16*8)), block = 256 (8 waves); one 16x16 tile per wave.
// ---------------------------------------------------------------------------
__global__ __launch_bounds__(256)
void proj_kernel(const float* __restrict__ A, const float* __restrict__ W,
                 float* __restrict__ Out) {
  const int lane  = threadIdx.x & 31;
  const int wave  = threadIdx.x >> 5;
  const int mtile = blockIdx.x;
  const int ntile = blockIdx.y * 8 + wave;

  const int half  = lane >> 4;           // 0: lanes 0-15, 1: lanes 16-31
  const int l16   = lane & 15;
  const int koff  = half * 2;            // A/B fragment K sub-offset
  const int row   = mtile * 16 + l16;    // A row for this lane
  const int col   = ntile * 16 + l16;    // B/C column for this lane

  v8f c = {};
  #pragma unroll 4
  for (int k0 = 0; k0 < DD; k0 += 4) {
    v2f a, b;
    a.x = A[(size_t)row * DD + k0 + koff];
    a.y = A[(size_t)row * DD + k0 + koff + 1];
    b.x = W[(size_t)(k0 + koff) * HH + col];
    b.y = W[(size_t)(k0 + koff + 1) * HH + col];
    c = __builtin_amdgcn_wmma_f32_16x16x4_f32(false, a, false, b,
                                              (short)0, c, false, false);
  }
  #pragma unroll
  for (int r = 0; r < 8; ++r) {
    const int orow = mtile * 16 + r + half * 8;
    Out[(size_t)orow * HH + col] = c[r];
  }
}

// ---------------------------------------------------------------------------
// Kernel 2: fused scores(tanh) + masked softmax + attn@V
// grid = B*(Q/16) blocks, block = 256 (8 waves).
// ---------------------------------------------------------------------------
__global__ __launch_bounds__(256)
void attn_kernel(const float* __restrict__ qp,   // [B*Q, H]
                 const float* __restrict__ kp,   // [B*K, H]
                 const float* __restrict__ V,    // [B, K, D]
                 const int*   __restrict__ valid_lens, // [B]
                 const float* __restrict__ wv,   // [H]
                 float* __restrict__ out) {      // [B, Q, D]
  __shared__ float s_qp[16][HH];      // 32 KB: projected query tile
  __shared__ float s_wv[HH];          // 2 KB
  __shared__ float s_sc[16][KK + 1];  // scores -> attn weights (pad: no bank conflicts)

  const int tid   = threadIdx.x;
  const int lane  = tid & 31;
  const int wave  = tid >> 5;
  const int bq    = blockIdx.x;       // 0 .. B*(Q/16)-1
  const int b     = bq >> 3;          // Q/16 == 8 tiles per batch
  const int qtile = bq & 7;
  const int qbase = b * QQ + qtile * 16;   // row into qp

  // Stage q-tile and wv into LDS
  for (int i = tid; i < 16 * HH; i += 256)
    s_qp[i >> 9][i & (HH - 1)] = qp[(size_t)(qbase + (i >> 9)) * HH + (i & (HH - 1))];
  for (int i = tid; i < HH; i += 256) s_wv[i] = wv[i];
  __syncthreads();

  // ---- Stage 1: scores[q][k] = sum_h wv[h] * tanh(qp[q][h] + kp[k][h]) ----
  // 2048 (q,k) pairs; one pair per wave per step, lanes parallel over H.
  const float* kpb = kp + (size_t)b * KK * HH;
  for (int i = 0; i < 256; ++i) {
    const int p  = i * 8 + wave;      // pair index, ordered k-major for kp reuse
    const int q  = p & 15;
    const int kk = p >> 4;
    const float* krow = kpb + (size_t)kk * HH;
    float acc = 0.f;
    #pragma unroll 4
    for (int hh = 0; hh < HH; hh += 32) {
      const int h = hh + lane;
      acc = fmaf(s_wv[h], fast_tanh(s_qp[q][h] + krow[h]), acc);
    }
    #pragma unroll
    for (int off = 16; off; off >>= 1) acc += __shfl_xor(acc, off, 32);
    if (lane == 0) s_sc[q][kk] = acc;
  }
  __syncthreads();

  // ---- Stage 2: masked softmax over k; wave w handles rows w and w+8 ----
  const int len = valid_lens[b];
  #pragma unroll
  for (int rr = 0; rr < 2; ++rr) {
    const int r = wave + rr * 8;
    float e[4];
    float vmax = -3.4e38f;
    #pragma unroll
    for (int j = 0; j < 4; ++j) {
      const int cidx = lane + j * 32;
      const float s = (cidx < len) ? s_sc[r][cidx] : NEGBIG;
      e[j] = s;
      vmax = fmaxf(vmax, s);
    }
    #pragma unroll
    for (int off = 16; off; off >>= 1) vmax = fmaxf(vmax, __shfl_xor(vmax, off, 32));
    float sum = 0.f;
    #pragma unroll
    for (int j = 0; j < 4; ++j) {
      e[j] = __builtin_amdgcn_exp2f((e[j] - vmax) * LOG2E);
      sum += e[j];
    }
    #pragma unroll
    for (int off = 16; off; off >>= 1) sum += __shfl_xor(sum, off, 32);
    const float rinv = 1.f / sum;
    #pragma unroll
    for (int j = 0; j < 4; ++j) s_sc[r][lane + j * 32] = e[j] * rinv;
  }
  __syncthreads();

  // ---- Stage 3: out[16, D] = attn[16, K] @ V[K, D] via fp32 WMMA ----
  const float* Vb  = V + (size_t)b * KK * DD;
  const int half = lane >> 4;
  const int l16  = lane & 15;
  const int koff = half * 2;
  #pragma unroll
  for (int t = 0; t < 4; ++t) {
    const int ntile = wave * 4 + t;           // 8 waves * 4 = 32 tiles of 16 -> D=512
    const int col   = ntile * 16 + l16;
    v8f c = {};
    #pragma unroll 4
    for (int k0 = 0; k0 < KK; k0 += 4) {
      v2f a, bf;
      a.x  = s_sc[l16][k0 + koff];
      a.y  = s_sc[l16][k0 + koff + 1];
      bf.x = Vb[(size_t)(k0 + koff) * DD + col];
      bf.y = Vb[(size_t)(k0 + koff + 1) * DD + col];
      c = __builtin_amdgcn_wmma_f32_16x16x4_f32(false, a, false, bf,
                                                (short)0, c, false, false);
    }
    #pragma unroll
    for (int r = 0; r < 8; ++r) {
      const int orow = qtile * 16 + r + half * 8;  // q-row within batch
      out[((size_t)b * QQ + orow) * DD + col] = c[r];
    }
  }
}

extern "C" void kernel_launch(void* const* d_in, const int* in_sizes, int n_in,
                              void* d_out, int out_size, void* d_ws, size_t ws_size,
                              hipStream_t stream) {
  const float* queries    = (const float*)d_in[0]; // [B,Q,D]
  const float* keys       = (const float*)d_in[1]; // [B,K,D]
  const float* values     = (const float*)d_in[2]; // [B,K,D]
  const int*   valid_lens = (const int*)  d_in[3]; // [B]
  const float* Wq         = (const float*)d_in[4]; // [D,H]
  const float* Wk         = (const float*)d_in[5]; // [D,H]
  const float* wv         = (const float*)d_in[6]; // [H]
  float* out = (float*)d_out;                      // [B,Q,D]

  float* qp = (float*)d_ws;                        // [B*Q, H] = 4 MB
  float* kp = qp + (size_t)BB * QQ * HH;           // [B*K, H] = 4 MB

  // Projections: M = B*Q = B*K = 2048 rows, N = H = 512
  dim3 pg(BB * QQ / 16, HH / (16 * 8));
  proj_kernel<<<pg, 256, 0, stream>>>(queries, Wq, qp);
  proj_kernel<<<pg, 256, 0, stream>>>(keys,    Wk, kp);

  // Fused scores + softmax + attn@V
  attn_kernel<<<BB * (QQ / 16), 256, 0, stream>>>(qp, kp, values, valid_lens, wv, out);
}